// Cell_71700184039583
// MI455X (gfx1250) — compile-verified
//
#include <hip/hip_runtime.h>
#include <math.h>

typedef __attribute__((ext_vector_type(2))) float v2f;
typedef __attribute__((ext_vector_type(8))) float v8f;

#define N_NODES 20000
#define N_EDGES 320000
#define DIM     256

__device__ __forceinline__ void atomic_add_f32(float* p, float v) {
  __hip_atomic_fetch_add(p, v, __ATOMIC_RELAXED, __HIP_MEMORY_SCOPE_AGENT);
}

// ---------------------------------------------------------------------------
// h = x @ W + b using V_WMMA_F32_16X16X4_F32. One wave per 16x16 output tile.
// A-frag (16x4 f32): lanes 0-15 hold K=k+0/k+1, lanes 16-31 hold K=k+2/k+3.
// B-frag (4x16 f32): mirrored (N across lanes). C/D: 8 VGPRs, row-striped.
// ---------------------------------------------------------------------------
__global__ __launch_bounds__(256) void gemm_xw_b(
    const float* __restrict__ x, const float* __restrict__ W,
    const float* __restrict__ bias, float* __restrict__ h) {
  const int wave = (blockIdx.x * blockDim.x + threadIdx.x) >> 5;
  const int lane = threadIdx.x & 31;
  const int tiles_n = DIM / 16;                 // 16 col tiles
  const int tm = wave / tiles_n;                // row tile (0..1249)
  const int tn = wave % tiles_n;                // col tile (0..15)
  if (tm * 16 >= N_NODES) return;               // wave-uniform

  const int hi   = lane >> 4;                   // 0: lanes 0-15, 1: lanes 16-31
  const int row  = tm * 16 + (lane & 15);       // A row for this lane
  const int col  = tn * 16 + (lane & 15);       // B/D col for this lane
  const int kq   = hi * 2;                      // K sub-offset (0 or 2)

  v8f acc = {};
  const float* xr = x + (size_t)row * DIM;
  #pragma unroll 4
  for (int k = 0; k < DIM; k += 4) {
    v2f a, b;
    a.x = xr[k + kq];
    a.y = xr[k + kq + 1];
    b.x = W[(size_t)(k + kq)     * DIM + col];
    b.y = W[(size_t)(k + kq + 1) * DIM + col];
    acc = __builtin_amdgcn_wmma_f32_16x16x4_f32(
        /*neg_a=*/false, a, /*neg_b=*/false, b,
        /*c_mod=*/(short)0, acc, /*reuse_a=*/false, /*reuse_b=*/false);
  }

  const float bv = bias[col];
  const int rbase = tm * 16 + hi * 8;           // D rows: VGPR r -> M = rbase+r
  #pragma unroll
  for (int r = 0; r < 8; ++r)
    h[(size_t)(rbase + r) * DIM + col] = acc[r] + bv;
}

// ---------------------------------------------------------------------------
// Edge-parallel SpMM passes. blockIdx.y = adjacency. 64 threads per edge
// (one float4 of the 256-wide feature row each) -> whole wave shares one edge.
// Each pass gathers src[col] once and scatters into up to 3 accumulators with
// precombined coefficients (weights / op-arity).
// ---------------------------------------------------------------------------
__global__ __launch_bounds__(256) void spmm_passA(
    const float* __restrict__ h,
    const int* __restrict__ rows, const int* __restrict__ cols,
    const float* __restrict__ vals,
    const float* __restrict__ ws_seq_0,   // [2,3]
    const float* __restrict__ ws_res_0,   // [1,4]
    const float* __restrict__ ws_res_1,   // [2,3]
    float* __restrict__ s1, float* __restrict__ s2, float* __restrict__ out) {
  const int a = blockIdx.y;
  const long long t = (long long)blockIdx.x * blockDim.x + threadIdx.x;
  const int e = (int)(t >> 6);              // edge index
  const int f = ((int)t & 63) * 4;          // feature offset
  if (e >= N_EDGES) return;

  // s1 = op(h,[0,1,2], ws_seq_0[0]) /3 ; s2 += op(h,[0..3], ws_res_0[0]) /4
  // out += op(h,[0,1,3], ws_res_1[0]) /3
  const float c_s1 = (a < 3) ? ws_seq_0[a] * (1.0f / 3.0f) : 0.0f;
  const float c_s2 = ws_res_0[a] * 0.25f;
  float c_out = 0.0f;
  if (a == 0) c_out = ws_res_1[0] * (1.0f / 3.0f);
  else if (a == 1) c_out = ws_res_1[1] * (1.0f / 3.0f);
  else if (a == 3) c_out = ws_res_1[2] * (1.0f / 3.0f);

  const size_t eidx = (size_t)a * N_EDGES + e;
  const int   r = rows[eidx];
  const int   c = cols[eidx];
  const float v = vals[eidx];

  const float4 hv = *(const float4*)(h + (size_t)c * DIM + f);
  const size_t ro = (size_t)r * DIM + f;
  if (c_s1 != 0.0f) {
    const float w = c_s1 * v;
    atomic_add_f32(s1 + ro + 0, w * hv.x); atomic_add_f32(s1 + ro + 1, w * hv.y);
    atomic_add_f32(s1 + ro + 2, w * hv.z); atomic_add_f32(s1 + ro + 3, w * hv.w);
  }
  {
    const float w = c_s2 * v;
    atomic_add_f32(s2 + ro + 0, w * hv.x); atomic_add_f32(s2 + ro + 1, w * hv.y);
    atomic_add_f32(s2 + ro + 2, w * hv.z); atomic_add_f32(s2 + ro + 3, w * hv.w);
  }
  if (c_out != 0.0f) {
    const float w = c_out * v;
    atomic_add_f32(out + ro + 0, w * hv.x); atomic_add_f32(out + ro + 1, w * hv.y);
    atomic_add_f32(out + ro + 2, w * hv.z); atomic_add_f32(out + ro + 3, w * hv.w);
  }
}

__global__ __launch_bounds__(256) void spmm_passB(
    const float* __restrict__ s1,
    const int* __restrict__ rows, const int* __restrict__ cols,
    const float* __restrict__ vals,
    const float* __restrict__ ws_seq_0,   // [2,3] (row 1 used)
    const float* __restrict__ ws_res_1,   // [2,3] (row 1 used)
    float* __restrict__ s2, float* __restrict__ out) {
  const int a = blockIdx.y;
  const long long t = (long long)blockIdx.x * blockDim.x + threadIdx.x;
  const int e = (int)(t >> 6);
  const int f = ((int)t & 63) * 4;
  if (e >= N_EDGES) return;

  // s2 += op(s1,[0,1,2], ws_seq_0[1]) /3 ; out += op(s1,[0,1,3], ws_res_1[1]) /3
  const float c_s2 = (a < 3) ? ws_seq_0[3 + a] * (1.0f / 3.0f) : 0.0f;
  float c_out = 0.0f;
  if (a == 0) c_out = ws_res_1[3] * (1.0f / 3.0f);
  else if (a == 1) c_out = ws_res_1[4] * (1.0f / 3.0f);
  else if (a == 3) c_out = ws_res_1[5] * (1.0f / 3.0f);

  const size_t eidx = (size_t)a * N_EDGES + e;
  const int   r = rows[eidx];
  const int   c = cols[eidx];
  const float v = vals[eidx];

  const float4 hv = *(const float4*)(s1 + (size_t)c * DIM + f);
  const size_t ro = (size_t)r * DIM + f;
  if (c_s2 != 0.0f) {
    const float w = c_s2 * v;
    atomic_add_f32(s2 + ro + 0, w * hv.x); atomic_add_f32(s2 + ro + 1, w * hv.y);
    atomic_add_f32(s2 + ro + 2, w * hv.z); atomic_add_f32(s2 + ro + 3, w * hv.w);
  }
  if (c_out != 0.0f) {
    const float w = c_out * v;
    atomic_add_f32(out + ro + 0, w * hv.x); atomic_add_f32(out + ro + 1, w * hv.y);
    atomic_add_f32(out + ro + 2, w * hv.z); atomic_add_f32(out + ro + 3, w * hv.w);
  }
}

__global__ __launch_bounds__(256) void spmm_passC(
    const float* __restrict__ s2,
    const int* __restrict__ rows, const int* __restrict__ cols,
    const float* __restrict__ vals,
    const float* __restrict__ ws_seq_1,   // [2]
    float* __restrict__ out) {
  const int a = blockIdx.y;                // 0 or 1
  const long long t = (long long)blockIdx.x * blockDim.x + threadIdx.x;
  const int e = (int)(t >> 6);
  const int f = ((int)t & 63) * 4;
  if (e >= N_EDGES) return;

  // out += op(s2,[0,1], ws_seq_1) /2
  const float c_out = ws_seq_1[a] * 0.5f;

  const size_t eidx = (size_t)a * N_EDGES + e;
  const int   r = rows[eidx];
  const int   c = cols[eidx];
  const float v = vals[eidx];

  const float4 hv = *(const float4*)(s2 + (size_t)c * DIM + f);
  const size_t ro = (size_t)r * DIM + f;
  const float w = c_out * v;
  atomic_add_f32(out + ro + 0, w * hv.x); atomic_add_f32(out + ro + 1, w * hv.y);
  atomic_add_f32(out + ro + 2, w * hv.z); atomic_add_f32(out + ro + 3, w * hv.w);
}

// ---------------------------------------------------------------------------
// LayerNorm (no affine) + exact GELU, in place. One row per 256-thread block.
// ---------------------------------------------------------------------------
__global__ __launch_bounds__(256) void ln_gelu(float* __restrict__ out) {
  const int row = blockIdx.x;
  const int t   = threadIdx.x;
  float v = out[(size_t)row * DIM + t];

  float s  = v;
  float ss = v * v;
  #pragma unroll
  for (int o = 16; o > 0; o >>= 1) {
    s  += __shfl_down(s, o, 32);
    ss += __shfl_down(ss, o, 32);
  }
  __shared__ float sh[16];
  const int w = t >> 5, l = t & 31;
  if (l == 0) { sh[w] = s; sh[8 + w] = ss; }
  __syncthreads();
  if (t == 0) {
    float a0 = 0.f, a1 = 0.f;
    #pragma unroll
    for (int i = 0; i < 8; ++i) { a0 += sh[i]; a1 += sh[8 + i]; }
    sh[0] = a0; sh[1] = a1;
  }
  __syncthreads();
  const float mean = sh[0] * (1.0f / DIM);
  const float var  = sh[1] * (1.0f / DIM) - mean * mean;
  const float nv   = (v - mean) * rsqrtf(var + 1e-5f);
  const float g    = 0.5f * nv * (1.0f + erff(nv * 0.70710678118654752440f));
  out[(size_t)row * DIM + t] = g;
}

extern "C" void kernel_launch(void* const* d_in, const int* in_sizes, int n_in,
                              void* d_out, int out_size, void* d_ws, size_t ws_size,
                              hipStream_t stream) {
  const float* x        = (const float*)d_in[0];
  const int*   rows     = (const int*)  d_in[1];
  const int*   cols     = (const int*)  d_in[2];
  const float* vals     = (const float*)d_in[3];
  const float* W        = (const float*)d_in[4];
  const float* bias     = (const float*)d_in[5];
  const float* ws_seq_0 = (const float*)d_in[6];
  const float* ws_seq_1 = (const float*)d_in[7];
  const float* ws_res_0 = (const float*)d_in[8];
  const float* ws_res_1 = (const float*)d_in[9];
  float* out = (float*)d_out;

  const size_t mat = (size_t)N_NODES * DIM;     // 5,120,000 floats
  float* h  = (float*)d_ws;
  float* s1 = h  + mat;
  float* s2 = s1 + mat;

  // Zero accumulators each call (graph-capturable memsets).
  hipMemsetAsync(s1, 0, 2 * mat * sizeof(float), stream);   // s1 and s2
  hipMemsetAsync(out, 0, mat * sizeof(float), stream);

  // 1) h = x @ W + b  (WMMA f32): 1250*16 tiles = 20000 waves = 2500 blocks.
  gemm_xw_b<<<2500, 256, 0, stream>>>(x, W, bias, h);

  // 2) SpMM passes: 320000 edges * 64 threads/edge / 256 = 80000 blocks per adj.
  const int eb = (N_EDGES * 64) / 256;
  spmm_passA<<<dim3(eb, 4), 256, 0, stream>>>(h, rows, cols, vals,
                                              ws_seq_0, ws_res_0, ws_res_1,
                                              s1, s2, out);
  spmm_passB<<<dim3(eb, 4), 256, 0, stream>>>(s1, rows, cols, vals,
                                              ws_seq_0, ws_res_1, s2, out);
  spmm_passC<<<dim3(eb, 2), 256, 0, stream>>>(s2, rows, cols, vals,
                                              ws_seq_1, out);

  // 3) LayerNorm + exact GELU in place on d_out.
  ln_gelu<<<N_NODES, 256, 0, stream>>>(out);
}